// VQAbstractEncoder_515396076053
// MI455X (gfx1250) — compile-verified
//
#include <hip/hip_runtime.h>

typedef float v2f __attribute__((ext_vector_type(2)));
typedef float v8f __attribute__((ext_vector_type(8)));

#define N_ROWS 32768
#define IN_DIM 256
#define EMB 64
#define NCODE 4096
#define LDS_STRIDE 68   // 64 + pad: lane*68 dwords -> conflict-free b64 reads

// ---------------- codebook norms: ||e_j||^2 ----------------
__global__ void enorm_kernel(const float* __restrict__ embed,
                             float* __restrict__ norms) {
  int j = blockIdx.x * blockDim.x + threadIdx.x;
  if (j < NCODE) {
    const float4* row = reinterpret_cast<const float4*>(embed + (size_t)j * EMB);
    float s = 0.f;
#pragma unroll
    for (int i = 0; i < EMB / 4; ++i) {
      float4 v = row[i];
      s += v.x * v.x + v.y * v.y + v.z * v.z + v.w * v.w;
    }
    norms[j] = s;
  }
}

// ---------------- fused body GEMM + VQ nearest-code search ----------------
__global__ __launch_bounds__(128) void vq_kernel(
    const float* __restrict__ in,     // [N_ROWS, IN_DIM]
    const float* __restrict__ W,      // [IN_DIM, EMB]
    const float* __restrict__ bias,   // [EMB]
    const float* __restrict__ embed,  // [NCODE, EMB]
    const float* __restrict__ norms,  // [NCODE]
    float* __restrict__ out) {        // [N_ROWS, EMB]
  __shared__ float xs_lds[4 * 16 * LDS_STRIDE];

  const int lane = threadIdx.x & 31;
  const int wave = threadIdx.x >> 5;
  const int l16  = lane & 15;       // A-row / B-col / D-col within fragment
  const int h    = lane >> 4;       // lane-half selects K pair / M upper half
  const int R    = (blockIdx.x * 4 + wave) * 16;  // 16-row tile base
  float* lds = xs_lds + wave * 16 * LDS_STRIDE;

  // ===== Phase 1: xs[16x64] = in[16x256] @ W[256x64] + b =====
  v8f acc[4];
#pragma unroll
  for (int c = 0; c < 4; ++c) acc[c] = (v8f)0.f;

  const float* inRow = in + (size_t)(R + l16) * IN_DIM + 2 * h;
#pragma unroll 4
  for (int s = 0; s < IN_DIM / 4; ++s) {
    // A fragment: rows across lanes, K = 4s + 2h + {0,1}
    v2f a = *reinterpret_cast<const v2f*>(inRow + 4 * s);
    int k = 4 * s + 2 * h;
#pragma unroll
    for (int c = 0; c < 4; ++c) {
      v2f b;
      b.x = W[(size_t)k * EMB + 16 * c + l16];
      b.y = W[(size_t)(k + 1) * EMB + 16 * c + l16];
      acc[c] = __builtin_amdgcn_wmma_f32_16x16x4_f32(
          false, a, false, b, (short)0, acc[c], false, false);
    }
  }

  // D-layout -> row-major xs tile in LDS, fusing the bias add
#pragma unroll
  for (int c = 0; c < 4; ++c) {
    float bb = bias[16 * c + l16];
#pragma unroll
    for (int r = 0; r < 8; ++r) {
      lds[(r + 8 * h) * LDS_STRIDE + 16 * c + l16] = acc[c][r] + bb;
    }
  }
  __syncthreads();

  // Reload xs as 16 A-fragments covering K = 0..63
  v2f afrag[16];
#pragma unroll
  for (int s = 0; s < 16; ++s)
    afrag[s] = *reinterpret_cast<const v2f*>(&lds[l16 * LDS_STRIDE + 4 * s + 2 * h]);

  // ===== Phase 2: scan 4096 codes, track argmin(||e||^2 - 2 xs.e) =====
  float best[8];
  int   bidx[8];
#pragma unroll
  for (int r = 0; r < 8; ++r) { best[r] = 3.402823466e38f; bidx[r] = 0; }

  for (int t = 0; t < NCODE / 16; ++t) {
    const int j = 16 * t + l16;                    // this lane's code column
    const float* erow = embed + (size_t)j * EMB + 2 * h;
    v2f bfrag[16];
#pragma unroll
    for (int s = 0; s < 16; ++s)                   // E^T fragments: K = 4s+2h+{0,1}
      bfrag[s] = *reinterpret_cast<const v2f*>(erow + 4 * s);

    v8f d = (v8f)0.f;
#pragma unroll
    for (int s = 0; s < 16; ++s)
      d = __builtin_amdgcn_wmma_f32_16x16x4_f32(
          false, afrag[s], false, bfrag[s], (short)0, d, false, false);

    float en = norms[j];
#pragma unroll
    for (int r = 0; r < 8; ++r) {
      float dist = en - 2.f * d[r];                // ||xs||^2 omitted: row-constant
      if (dist < best[r]) { best[r] = dist; bidx[r] = j; }
    }
  }

  // ===== Butterfly min-reduce across the 16 code-lanes, then gather =====
#pragma unroll
  for (int r = 0; r < 8; ++r) {
    float bv = best[r];
    int   bi = bidx[r];
#pragma unroll
    for (int off = 8; off > 0; off >>= 1) {
      float ov = __shfl_xor(bv, off, 16);
      int   oi = __shfl_xor(bi, off, 16);
      if (ov < bv || (ov == bv && oi < bi)) { bv = ov; bi = oi; }
    }
    // half h holds the result for row r + 8h; 16 lanes stream the code row
    int row = R + r + 8 * h;
    const float4* src = reinterpret_cast<const float4*>(embed + (size_t)bi * EMB);
    float4*       dst = reinterpret_cast<float4*>(out + (size_t)row * EMB);
    dst[l16] = src[l16];
  }
}

extern "C" void kernel_launch(void* const* d_in, const int* in_sizes, int n_in,
                              void* d_out, int out_size, void* d_ws, size_t ws_size,
                              hipStream_t stream) {
  const float* in    = (const float*)d_in[0];   // [32768,256]
  const float* W     = (const float*)d_in[1];   // [256,64]
  const float* bias  = (const float*)d_in[2];   // [64]
  const float* embed = (const float*)d_in[3];   // [4096,64]
  float* norms = (float*)d_ws;                  // 4096 floats scratch
  float* out   = (float*)d_out;                 // [32768,64]

  enorm_kernel<<<NCODE / 256, 256, 0, stream>>>(embed, norms);
  vq_kernel<<<N_ROWS / 64, 128, 0, stream>>>(in, W, bias, embed, norms, out);
}